// GHMC_45810121179416
// MI455X (gfx1250) — compile-verified
//
#include <hip/hip_runtime.h>
#include <stdint.h>

#define BINS 10
#define BLOCK 256
#define WAVES_PER_BLOCK (BLOCK / 32)
#define TILE_ELEMS 128              // per wave per tile: 32 lanes * 4 elements
#define NBUF 4                      // async pipeline depth (tiles in flight)
#define NBLOCKS 1024

typedef __attribute__((ext_vector_type(2))) float v2f;
typedef __attribute__((ext_vector_type(8))) float v8f;

// ---------------------------------------------------------------------------
// CDNA5 async global->LDS copy (per-lane b128) + asynccnt wait
// ---------------------------------------------------------------------------
__device__ __forceinline__ void async_load_b128(uint32_t lds_byte_addr, const void* gaddr) {
    asm volatile("global_load_async_to_lds_b128 %0, %1, off"
                 :: "v"(lds_byte_addr), "v"((uint64_t)(uintptr_t)gaddr)
                 : "memory");
}

template <int N>
__device__ __forceinline__ void wait_asynccnt() {
#if __has_builtin(__builtin_amdgcn_s_wait_asynccnt)
    __builtin_amdgcn_s_wait_asynccnt(N);
    asm volatile("" ::: "memory");
#else
    asm volatile("s_wait_asynccnt %0" :: "n"(N) : "memory");
#endif
}

__device__ __forceinline__ unsigned ballot32(bool p) {
#if __has_builtin(__builtin_amdgcn_ballot_w32)
    return __builtin_amdgcn_ballot_w32(p);
#else
    return (unsigned)__ballot(p);
#endif
}

// ---------------------------------------------------------------------------
// Per-element GHMC accumulate.
//  - sums: per-lane f32 register bins (v_cmp + v_cndmask + v_add)
//  - counts: ballot -> SALU popcount/add (wave-uniform, co-executes with VALU)
// ---------------------------------------------------------------------------
__device__ __forceinline__ void acc_elem(float x, int t,
                                         float (&s)[BINS], unsigned (&cnt)[BINS]) {
    float pe   = 1.0f / (1.0f + __expf(-x));      // sigmoid
    float tf   = (float)t;
    float g    = fabsf(pe - tf);                  // in [0,1]
    int   idx  = (int)(g * 10.0f);                // floor (g >= 0); g==1 -> 10 (no bin)
    float larg = t ? pe : (1.0f - pe);            // t in {0,1}: only one log needed
    float bce  = -fmaxf(__logf(larg), -100.0f);   // BCE with torch -100 clamp
#pragma unroll
    for (int b = 0; b < BINS; ++b) {
        bool m = (idx == b);
        s[b]   += m ? bce : 0.0f;
        cnt[b] += (unsigned)__popc(ballot32(m));  // wave-total, uniform across lanes
    }
}

// ---------------------------------------------------------------------------
// Kernel 0: zero workspace + output (workspace is poisoned; must init per call)
// ---------------------------------------------------------------------------
__global__ void ghmc_zero(float* wsSums, unsigned* wsCnts, float* out) {
    if (threadIdx.x < BINS) {
        wsSums[threadIdx.x] = 0.0f;
        wsCnts[threadIdx.x] = 0u;
    }
    if (threadIdx.x == 0) out[0] = 0.0f;
}

// ---------------------------------------------------------------------------
// Kernel 1: single streaming pass. 4-deep async global->LDS pipeline,
// per-lane f32 register bins + SALU ballot counts, per-wave LDS reduce,
// per-block global atomics.
// ---------------------------------------------------------------------------
__global__ __launch_bounds__(BLOCK) void ghmc_main(const float* __restrict__ pred,
                                                   const int*   __restrict__ targ,
                                                   int n,
                                                   float* __restrict__ wsSums,
                                                   unsigned* __restrict__ wsCnts) {
    __shared__ __align__(16) float sPred[NBUF][WAVES_PER_BLOCK][TILE_ELEMS];
    __shared__ __align__(16) int   sTarg[NBUF][WAVES_PER_BLOCK][TILE_ELEMS];
    __shared__ float    sBin[WAVES_PER_BLOCK][BINS];
    __shared__ unsigned cBin[WAVES_PER_BLOCK][BINS];

    const int lane       = threadIdx.x & 31;
    const int waveInBlk  = threadIdx.x >> 5;
    const int waveGlobal = blockIdx.x * WAVES_PER_BLOCK + waveInBlk;
    const int numWaves   = gridDim.x * WAVES_PER_BLOCK;

    // init this wave's private sum bins (only this wave touches them)
    if (lane < BINS) sBin[waveInBlk][lane] = 0.0f;

    float    s[BINS];
    unsigned cnt[BINS];
#pragma unroll
    for (int b = 0; b < BINS; ++b) { s[b] = 0.0f; cnt[b] = 0u; }

    const int nTiles = n / TILE_ELEMS;

    uint32_t ldsP[NBUF], ldsT[NBUF];
#pragma unroll
    for (int bf = 0; bf < NBUF; ++bf) {
        ldsP[bf] = (uint32_t)(uintptr_t)&sPred[bf][waveInBlk][lane * 4];
        ldsT[bf] = (uint32_t)(uintptr_t)&sTarg[bf][waveInBlk][lane * 4];
    }

    // prologue: issue tiles k = 0 .. NBUF-2 ahead (wave-uniform branches)
#pragma unroll
    for (int k = 0; k < NBUF - 1; ++k) {
        int tk = waveGlobal + k * numWaves;
        if (tk < nTiles) {
            async_load_b128(ldsP[k], pred + (size_t)tk * TILE_ELEMS + lane * 4);
            async_load_b128(ldsT[k], targ + (size_t)tk * TILE_ELEMS + lane * 4);
        }
    }

    int k = 0;
    for (int tile = waveGlobal; tile < nTiles; tile += numWaves, ++k) {
        const int issueTile = tile + (NBUF - 1) * numWaves;
        const int buf       = k & (NBUF - 1);
        if (issueTile < nTiles) {
            const int ibuf = (k + NBUF - 1) & (NBUF - 1);
            async_load_b128(ldsP[ibuf], pred + (size_t)issueTile * TILE_ELEMS + lane * 4);
            async_load_b128(ldsT[ibuf], targ + (size_t)issueTile * TILE_ELEMS + lane * 4);
            wait_asynccnt<2 * (NBUF - 1)>();   // oldest tile's 2 loads complete (in-order)
        } else {
            wait_asynccnt<0>();                // drain
        }
        float4 p4 = *(const float4*)&sPred[buf][waveInBlk][lane * 4];
        int4   t4 = *(const int4*)  &sTarg[buf][waveInBlk][lane * 4];
        acc_elem(p4.x, t4.x, s, cnt);
        acc_elem(p4.y, t4.y, s, cnt);
        acc_elem(p4.z, t4.z, s, cnt);
        acc_elem(p4.w, t4.w, s, cnt);
    }

    // tail (n not a multiple of TILE_ELEMS): plain loads, divergence-safe
    for (int i = nTiles * TILE_ELEMS + blockIdx.x * BLOCK + threadIdx.x;
         i < n; i += gridDim.x * BLOCK) {
        acc_elem(pred[i], targ[i], s, cnt);
    }

    // per-wave reduce: sums via ds_add_f32 atomics; counts are wave-uniform
#pragma unroll
    for (int b = 0; b < BINS; ++b) atomicAdd(&sBin[waveInBlk][b], s[b]);
    if (lane == 0) {
#pragma unroll
        for (int b = 0; b < BINS; ++b) cBin[waveInBlk][b] = cnt[b];
    }
    __syncthreads();

    // per-block: combine wave replicas, one global atomic per bin
    if (threadIdx.x < BINS) {
        float    fs = 0.0f;
        unsigned fc = 0u;
        for (int w = 0; w < WAVES_PER_BLOCK; ++w) {
            fs += sBin[w][threadIdx.x];
            fc += cBin[w][threadIdx.x];
        }
        atomicAdd(&wsSums[threadIdx.x], fs);
        atomicAdd(&wsCnts[threadIdx.x], fc);
    }
}

// ---------------------------------------------------------------------------
// Kernel 2: finalize.  result = (1/n_nonempty) * sum_b S_b / max(c_b, 1)
// The 10-term sum is computed with 3 chained V_WMMA_F32_16X16X4_F32:
// A row 0 carries 4 terms (lanes 0/16, K=0..3 per ISA layout), B = ones,
// C-accumulate chain; D[0][0] = lane 0, VGPR 0 of the accumulator.
// One wave, EXEC all ones, no divergence around the WMMAs.
// ---------------------------------------------------------------------------
__global__ __launch_bounds__(32) void ghmc_final(const float* __restrict__ wsSums,
                                                 const unsigned* __restrict__ wsCnts,
                                                 float* __restrict__ out) {
    const int lane = threadIdx.x;

    float t[12];
    int   nb = 0;
#pragma unroll
    for (int b = 0; b < BINS; ++b) {
        unsigned cb = wsCnts[b];
        nb += (cb > 0u) ? 1 : 0;
        t[b] = wsSums[b] / (float)((cb > 1u) ? cb : 1u);
    }
    t[10] = 0.0f;
    t[11] = 0.0f;

    v2f bmat;                       // B = ones(4x16): every element 1.0
    bmat.x = 1.0f;
    bmat.y = 1.0f;

    v8f acc = {};
#pragma unroll
    for (int ch = 0; ch < 3; ++ch) {
        const int base = ch * 4;
        v2f amat;                   // A[0][K]: lane0 -> K=0,1 ; lane16 -> K=2,3
        amat.x = (lane == 0) ? t[base + 0] : ((lane == 16) ? t[base + 2] : 0.0f);
        amat.y = (lane == 0) ? t[base + 1] : ((lane == 16) ? t[base + 3] : 0.0f);
        acc = __builtin_amdgcn_wmma_f32_16x16x4_f32(
            /*neg_a=*/false, amat, /*neg_b=*/false, bmat,
            /*c_mod=*/(short)0, acc, /*reuse_a=*/false, /*reuse_b=*/false);
    }

    const float total = acc.x;      // D[0][0] lives in lane 0
    if (lane == 0) out[0] = (nb > 0) ? (total / (float)nb) : 0.0f;
}

// ---------------------------------------------------------------------------
extern "C" void kernel_launch(void* const* d_in, const int* in_sizes, int n_in,
                              void* d_out, int out_size, void* d_ws, size_t ws_size,
                              hipStream_t stream) {
    const float* pred = (const float*)d_in[0];
    const int*   targ = (const int*)d_in[1];
    const int    n    = in_sizes[0];

    float*    wsSums = (float*)d_ws;
    unsigned* wsCnts = (unsigned*)((char*)d_ws + BINS * sizeof(float));
    float*    out    = (float*)d_out;

    ghmc_zero<<<1, 32, 0, stream>>>(wsSums, wsCnts, out);
    ghmc_main<<<NBLOCKS, BLOCK, 0, stream>>>(pred, targ, n, wsSums, wsCnts);
    ghmc_final<<<1, 32, 0, stream>>>(wsSums, wsCnts, out);
}